// PositionwiseMaxPool2D_19859928777087
// MI455X (gfx1250) — compile-verified
//
#include <hip/hip_runtime.h>
#include <stdint.h>

// PositionwiseMaxPool2D, 2x2/stride2, NHWC fp32.
// x:[32,128,128,64] -> out:[32,64,64,64]
// HBM-bound (~168MB/pass -> ~7.2us floor @ 23.3TB/s). Strategy: b128 streaming
// with CDNA5 async global->LDS double buffering (ASYNCcnt-tracked, barrier-free
// since each lane consumes exactly the LDS slots it staged), wave-local
// shfl_xor butterfly for the channel-norm reduction.

namespace {
constexpr int kB  = 32;
constexpr int kH  = 128;
constexpr int kW  = 128;
constexpr int kC  = 64;
constexpr int kOH = 64;
constexpr int kOW = 64;
constexpr int kTile  = 16;            // output pixels per tile
constexpr int kIters = kOW / kTile;   // 4 tiles per (b, oh) row
}

__global__ __launch_bounds__(256)
void pos_maxpool2d_kernel(const float* __restrict__ x, float* __restrict__ out) {
    // double buffer: [2][4 positions][16 pixels][64 channels] = 32 KB
    __shared__ __align__(16) float lds[2 * 4 * kTile * kC];

    const int tid = threadIdx.x;
    const int cg  = tid & 15;   // channel group: floats [4*cg .. 4*cg+3]
    const int pix = tid >> 4;   // pixel within tile: 0..15

    const int b   = blockIdx.x / kOH;
    const int ohh = blockIdx.x - b * kOH;

    // input row base for di=0 of this output row
    const float* __restrict__ xrow0 = x + ((size_t)(b * kH + 2 * ohh) * kW) * kC;
    // lane offset within an input row for dj=0 (pixel 2*pix, channel 4*cg)
    const int lane_off = (2 * pix) * kC + 4 * cg;

    // low 32 bits of a generic LDS pointer == LDS byte offset
    const uint32_t lds0 = (uint32_t)(uintptr_t)lds;

    // ---- prologue: stage tile 0 into buffer 0 ----
#pragma unroll
    for (int p = 0; p < 4; ++p) {
        const float* g = xrow0 + (p >> 1) * (kW * kC) + (p & 1) * kC + lane_off;
        const uint32_t l = lds0 + (uint32_t)(((0 * 4 + p) * kTile + pix) * kC + 4 * cg) * 4u;
        asm volatile("global_load_async_to_lds_b128 %0, %1, off"
                     :: "v"(l), "v"(g) : "memory");
    }

    float* __restrict__ obase =
        out + ((size_t)(b * kOH + ohh) * kOW) * kC + pix * kC + 4 * cg;

    for (int it = 0; it < kIters; ++it) {
        const int buf = it & 1;

        if (it + 1 < kIters) {
            // stage next tile into the other buffer, then wait so that the
            // first 4 outstanding (current tile, in-order completion) are done
            const int nbuf     = buf ^ 1;
            const int tile_off = 2 * (it + 1) * kTile * kC;  // ow0 shift in floats
#pragma unroll
            for (int p = 0; p < 4; ++p) {
                const float* g = xrow0 + (p >> 1) * (kW * kC) + (p & 1) * kC
                                       + lane_off + tile_off;
                const uint32_t l = lds0 +
                    (uint32_t)(((nbuf * 4 + p) * kTile + pix) * kC + 4 * cg) * 4u;
                asm volatile("global_load_async_to_lds_b128 %0, %1, off"
                             :: "v"(l), "v"(g) : "memory");
            }
            asm volatile("s_wait_asynccnt 0x4" ::: "memory");
        } else {
            asm volatile("s_wait_asynccnt 0x0" ::: "memory");
        }

        // load this lane's 4 candidate float4s back and form partial norms
        float4 v[4];
        float  n[4];
#pragma unroll
        for (int p = 0; p < 4; ++p) {
            const float4* lp = (const float4*)
                &lds[((buf * 4 + p) * kTile + pix) * kC + 4 * cg];
            v[p] = *lp;
            n[p] = v[p].x * v[p].x + v[p].y * v[p].y
                 + v[p].z * v[p].z + v[p].w * v[p].w;
        }

        // butterfly-sum the 4 norms over the 16 lanes sharing this pixel
#pragma unroll
        for (int m = 8; m >= 1; m >>= 1) {
#pragma unroll
            for (int p = 0; p < 4; ++p) n[p] += __shfl_xor(n[p], m, 32);
        }

        // first-max argmax (matches jnp.argmax tie-breaking)
        int   best = 0;
        float bn   = n[0];
        if (n[1] > bn) { bn = n[1]; best = 1; }
        if (n[2] > bn) { bn = n[2]; best = 2; }
        if (n[3] > bn) { bn = n[3]; best = 3; }

        float4 w = v[0];
        w = (best == 1) ? v[1] : w;
        w = (best == 2) ? v[2] : w;
        w = (best == 3) ? v[3] : w;

        *(float4*)(obase + it * kTile * kC) = w;  // global_store_b128
    }
}

extern "C" void kernel_launch(void* const* d_in, const int* in_sizes, int n_in,
                              void* d_out, int out_size, void* d_ws, size_t ws_size,
                              hipStream_t stream) {
    (void)in_sizes; (void)n_in; (void)out_size; (void)d_ws; (void)ws_size;
    const float* x = (const float*)d_in[0];
    float* out = (float*)d_out;

    dim3 grid(kB * kOH);   // 2048 blocks: one (batch, output-row) pair each
    dim3 block(256);       // 8 wave32; 16 pixels x 16 channel-groups
    pos_maxpool2d_kernel<<<grid, block, 0, stream>>>(x, out);
}